// Dependency_GCN_18098992185957
// MI455X (gfx1250) — compile-verified
//
#include <hip/hip_runtime.h>
#include <hip/hip_bf16.h>

#define NTOK 8192
#define DIM  512
#define NREL 20
#define OUTD 256
#define PAD_CAP (NTOK + NREL * 16)   // 8512 edge slots (each bucket padded to 16)
#define TILE_CAP (PAD_CAP / 16)      // 532 tiles upper bound

typedef __attribute__((ext_vector_type(2))) float v2f;
typedef __attribute__((ext_vector_type(8))) float v8f;

__device__ __forceinline__ void gatomic_add_f32(float* p, float v) {
    // fp32 add executed by the L2 atomic units; no return (STOREcnt path)
    asm volatile("global_atomic_add_f32 %0, %1, off"
                 :: "v"((unsigned long long)p), "v"(v)
                 : "memory");
}

__device__ __forceinline__ v8f wmma4(v2f a, v2f b, v8f c) {
    return __builtin_amdgcn_wmma_f32_16x16x4_f32(false, a, false, b, (short)0, c,
                                                 false, false);
}

// ---------------- edge bucketing (counting sort by relation) ----------------

__global__ void k_init(int* cnt, int* cursor, int* tile_group, int* elist) {
    int i = blockIdx.x * blockDim.x + threadIdx.x;
    if (i < NREL) { cnt[i] = 0; cursor[i] = 0; }
    if (i < TILE_CAP) tile_group[i] = -1;
    for (int j = i; j < PAD_CAP; j += gridDim.x * blockDim.x) elist[j] = -1;
}

__global__ void k_count(const int* __restrict__ rel, int* cnt) {
    int i = blockIdx.x * blockDim.x + threadIdx.x;
    if (i < NTOK) atomicAdd(&cnt[rel[i]], 1);
}

__global__ void k_scan(const int* __restrict__ cnt, int* off, int* tile_group) {
    // single thread: 20 buckets + <=532 tile tags
    off[0] = 0;
    for (int r = 0; r < NREL; ++r)
        off[r + 1] = off[r] + (((cnt[r] + 15) >> 4) << 4);
    for (int r = 0; r < NREL; ++r)
        for (int t = off[r] >> 4; t < (off[r + 1] >> 4); ++t)
            tile_group[t] = r;
}

__global__ void k_fill(const int* __restrict__ rel, const int* __restrict__ off,
                       int* cursor, int* elist) {
    int i = blockIdx.x * blockDim.x + threadIdx.x;
    if (i < NTOK) {
        int r = rel[i];
        int pos = off[r] + atomicAdd(&cursor[r], 1);
        elist[pos] = i;
    }
}

// ---------------- dense GEMM: C[M x Dout] = A[M x K] @ W[Dout x K]^T + bias --
// Register-blocked 32x32 per wave (4 accumulators), K-block double-buffered:
// loads for block k+16 are issued before the 16 WMMAs of block k execute.

__global__ __launch_bounds__(128) void k_gemm_bias(
    const float* __restrict__ A, const float* __restrict__ W,
    const float* __restrict__ bias, float* __restrict__ C,
    int K, int Dout) {
    const int lane = threadIdx.x & 31;
    const int wave = threadIdx.x >> 5;
    const int mrow = lane & 15;       // A: M index, B: N index, C: N index
    const int hi   = lane >> 4;       // K half-select for A/B, M half for C
    const int m0 = blockIdx.x * 64 + (wave >> 1) * 32;
    const int n0 = blockIdx.y * 64 + (wave & 1) * 32;

    const float* arow0 = A + (size_t)(m0 + mrow) * K + 2 * hi;
    const float* arow1 = arow0 + (size_t)16 * K;
    const float* wrow0 = W + (size_t)(n0 + mrow) * K + 2 * hi;
    const float* wrow1 = wrow0 + (size_t)16 * K;

    v2f aLo[4], aHi[4], bLo[4], bHi[4];      // current K-block (4 sub-steps)
    v2f naLo[4], naHi[4], nbLo[4], nbHi[4];  // next K-block

#pragma unroll
    for (int j = 0; j < 4; ++j) {
        aLo[j] = *(const v2f*)(arow0 + 4 * j);
        aHi[j] = *(const v2f*)(arow1 + 4 * j);
        bLo[j] = *(const v2f*)(wrow0 + 4 * j);
        bHi[j] = *(const v2f*)(wrow1 + 4 * j);
    }

    v8f c00 = {}, c01 = {}, c10 = {}, c11 = {};
    for (int k = 0; k < K - 16; k += 16) {
        // prefetch next block into the second buffer
#pragma unroll
        for (int j = 0; j < 4; ++j) {
            naLo[j] = *(const v2f*)(arow0 + k + 16 + 4 * j);
            naHi[j] = *(const v2f*)(arow1 + k + 16 + 4 * j);
            nbLo[j] = *(const v2f*)(wrow0 + k + 16 + 4 * j);
            nbHi[j] = *(const v2f*)(wrow1 + k + 16 + 4 * j);
        }
        // compute on current block while next block's loads are in flight
#pragma unroll
        for (int j = 0; j < 4; ++j) {
            c00 = wmma4(aLo[j], bLo[j], c00);
            c01 = wmma4(aLo[j], bHi[j], c01);
            c10 = wmma4(aHi[j], bLo[j], c10);
            c11 = wmma4(aHi[j], bHi[j], c11);
        }
#pragma unroll
        for (int j = 0; j < 4; ++j) {
            aLo[j] = naLo[j]; aHi[j] = naHi[j];
            bLo[j] = nbLo[j]; bHi[j] = nbHi[j];
        }
    }
    // epilogue: last K-block
#pragma unroll
    for (int j = 0; j < 4; ++j) {
        c00 = wmma4(aLo[j], bLo[j], c00);
        c01 = wmma4(aLo[j], bHi[j], c01);
        c10 = wmma4(aHi[j], bLo[j], c10);
        c11 = wmma4(aHi[j], bHi[j], c11);
    }

    const float bn0 = bias[n0 + mrow];
    const float bn1 = bias[n0 + 16 + mrow];
    const int mb0 = m0 + 8 * hi;
    const int mb1 = m0 + 16 + 8 * hi;
#pragma unroll
    for (int r2 = 0; r2 < 8; ++r2) {
        C[(size_t)(mb0 + r2) * Dout + n0 + mrow]      = c00[r2] + bn0;
        C[(size_t)(mb0 + r2) * Dout + n0 + 16 + mrow] = c01[r2] + bn1;
        C[(size_t)(mb1 + r2) * Dout + n0 + mrow]      = c10[r2] + bn0;
        C[(size_t)(mb1 + r2) * Dout + n0 + 16 + mrow] = c11[r2] + bn1;
    }
}

// ---- edge-message GEMM: 16 gathered rows x W_rel[g]^T, scatter-add to Out ---
// Each wave computes 16x32 (1 A-frag + 2 B-frags per step, double-buffered);
// block of 4 waves covers a 16x128 output slice.

__global__ __launch_bounds__(128) void k_edge(
    const float* __restrict__ H, const float* __restrict__ Wrel, // [2R][D][D]
    const float* __restrict__ brel,                              // [2R][D]
    float* __restrict__ Out,
    const int* __restrict__ dep, const int* __restrict__ gov,
    const int* __restrict__ tile_group, const int* __restrict__ elist) {
    __shared__ int s_src[16];
    __shared__ int s_dst[16];

    const int t = blockIdx.x;
    const int g = tile_group[t];          // uniform per block
    if (g < 0) return;
    const int dir = blockIdx.z;           // 0 = forward (gov->dep), 1 = reverse

    if (threadIdx.x < 16) {
        int e = elist[t * 16 + threadIdx.x];
        int s = 0, d = -1;
        if (e >= 0) {
            if (dir == 0) { s = gov[e]; d = dep[e]; }
            else          { s = dep[e]; d = gov[e]; }
        }
        s_src[threadIdx.x] = s;
        s_dst[threadIdx.x] = d;
    }
    __syncthreads();

    const int lane = threadIdx.x & 31;
    const int wave = threadIdx.x >> 5;
    const int mrow = lane & 15;
    const int hi   = lane >> 4;
    const int n0 = blockIdx.y * 128 + wave * 32;

    const int wsel = dir ? (NREL + g) : g;
    const float* Wm = Wrel + (size_t)wsel * DIM * DIM;
    const float* bm = brel + (size_t)wsel * DIM;

    const float* arow  = H + (size_t)s_src[mrow] * DIM + 2 * hi;
    const float* wrow0 = Wm + (size_t)(n0 + mrow) * DIM + 2 * hi;
    const float* wrow1 = wrow0 + (size_t)16 * DIM;

    v2f a[4], bLo[4], bHi[4];
    v2f na[4], nbLo[4], nbHi[4];
#pragma unroll
    for (int j = 0; j < 4; ++j) {
        a[j]   = *(const v2f*)(arow + 4 * j);
        bLo[j] = *(const v2f*)(wrow0 + 4 * j);
        bHi[j] = *(const v2f*)(wrow1 + 4 * j);
    }

    v8f c0 = {}, c1 = {};
    for (int k = 0; k < DIM - 16; k += 16) {
#pragma unroll
        for (int j = 0; j < 4; ++j) {
            na[j]   = *(const v2f*)(arow + k + 16 + 4 * j);
            nbLo[j] = *(const v2f*)(wrow0 + k + 16 + 4 * j);
            nbHi[j] = *(const v2f*)(wrow1 + k + 16 + 4 * j);
        }
#pragma unroll
        for (int j = 0; j < 4; ++j) {
            c0 = wmma4(a[j], bLo[j], c0);
            c1 = wmma4(a[j], bHi[j], c1);
        }
#pragma unroll
        for (int j = 0; j < 4; ++j) {
            a[j] = na[j]; bLo[j] = nbLo[j]; bHi[j] = nbHi[j];
        }
    }
#pragma unroll
    for (int j = 0; j < 4; ++j) {
        c0 = wmma4(a[j], bLo[j], c0);
        c1 = wmma4(a[j], bHi[j], c1);
    }

    const float bn0 = bm[n0 + mrow];
    const float bn1 = bm[n0 + 16 + mrow];
#pragma unroll
    for (int r2 = 0; r2 < 8; ++r2) {
        int d = s_dst[r2 + 8 * hi];
        if (d >= 0) {
            gatomic_add_f32(&Out[(size_t)d * DIM + n0 + mrow],      c0[r2] + bn0);
            gatomic_add_f32(&Out[(size_t)d * DIM + n0 + 16 + mrow], c1[r2] + bn1);
        }
    }
}

// ---------------- elementwise ReLU ------------------------------------------

__global__ void k_relu(float* __restrict__ p, int n) {
    int i = blockIdx.x * blockDim.x + threadIdx.x;
    if (i < n) {
        float v = p[i];
        p[i] = v > 0.0f ? v : 0.0f;
    }
}

// ---------------- host-side launch ------------------------------------------

extern "C" void kernel_launch(void* const* d_in, const int* in_sizes, int n_in,
                              void* d_out, int out_size, void* d_ws, size_t ws_size,
                              hipStream_t stream) {
    const float* x      = (const float*)d_in[0];
    const int*   dep    = (const int*)d_in[1];
    const int*   rel    = (const int*)d_in[2];
    const int*   gov    = (const int*)d_in[3];
    const float* W_self = (const float*)d_in[4]; // [2][512][512]
    const float* b_self = (const float*)d_in[5]; // [2][512]
    const float* W_rel  = (const float*)d_in[6]; // [2][40][512][512]
    const float* b_rel  = (const float*)d_in[7]; // [2][40][512]
    const float* W_ff   = (const float*)d_in[8]; // [256][512]
    const float* b_ff   = (const float*)d_in[9]; // [256]
    float* out = (float*)d_out;

    // workspace layout: [int meta | 64KB-aligned float buffers]
    int* cnt        = (int*)d_ws;
    int* cursor     = cnt + NREL;
    int* off        = cursor + NREL;           // 21 entries
    int* tile_group = off + (NREL + 1);        // TILE_CAP
    int* elist      = tile_group + TILE_CAP;   // PAD_CAP
    float* buf0 = (float*)((char*)d_ws + 64 * 1024);
    float* buf1 = buf0 + (size_t)NTOK * DIM;

    // --- build per-relation edge buckets (static across layers) ---
    k_init <<<34, 256, 0, stream>>>(cnt, cursor, tile_group, elist);
    k_count<<<NTOK / 256, 256, 0, stream>>>(rel, cnt);
    k_scan <<<1, 1, 0, stream>>>(cnt, off, tile_group);
    k_fill <<<NTOK / 256, 256, 0, stream>>>(rel, off, cursor, elist);

    dim3 gemm_grid(NTOK / 64, DIM / 64);
    dim3 edge_grid(TILE_CAP, DIM / 128, 2);
    const int relu_blocks = (NTOK * DIM) / 256;

    // --- layer 0: x -> buf0 ---
    k_gemm_bias<<<gemm_grid, 128, 0, stream>>>(x, W_self, b_self, buf0, DIM, DIM);
    k_edge<<<edge_grid, 128, 0, stream>>>(x, W_rel, b_rel, buf0, dep, gov,
                                          tile_group, elist);
    k_relu<<<relu_blocks, 256, 0, stream>>>(buf0, NTOK * DIM);

    // --- layer 1: buf0 -> buf1 ---
    const float* Ws1 = W_self + (size_t)DIM * DIM;
    const float* bs1 = b_self + DIM;
    const float* Wr1 = W_rel + (size_t)2 * NREL * DIM * DIM;
    const float* br1 = b_rel + (size_t)2 * NREL * DIM;
    k_gemm_bias<<<gemm_grid, 128, 0, stream>>>(buf0, Ws1, bs1, buf1, DIM, DIM);
    k_edge<<<edge_grid, 128, 0, stream>>>(buf0, Wr1, br1, buf1, dep, gov,
                                          tile_group, elist);
    k_relu<<<relu_blocks, 256, 0, stream>>>(buf1, NTOK * DIM);

    // --- final FF: buf1 -> out [N x 256] ---
    dim3 ff_grid(NTOK / 64, OUTD / 64);
    k_gemm_bias<<<ff_grid, 128, 0, stream>>>(buf1, W_ff, b_ff, out, DIM, OUTD);
}